// VQEmbedding_89232240541723
// MI455X (gfx1250) — compile-verified
//
#include <hip/hip_runtime.h>
#include <stdint.h>

typedef __attribute__((ext_vector_type(2))) float v2f;
typedef __attribute__((ext_vector_type(8))) float v8f;

#define N_EMB 1024
#define EMB_DIM 128
#define BATCH 8
#define SEQ 4096
#define NROWS (BATCH * SEQ)              // 32768
#define BCT (BATCH * EMB_DIM * SEQ)      // 4194304

#define CHUNK 64                          // codebook rows staged in LDS at once
#define LDS_STRIDE 132                    // padded row stride (floats): 528B = 33*16B, conflict-free
#define ROWS_PER_BLOCK 128                // 8 waves x 16 rows
#define TILES_PER_CHUNK (CHUNK / 16)      // 4
#define NCHUNK (N_EMB / CHUNK)            // 16

__device__ __forceinline__ void wait_asynccnt0() {
#if __has_builtin(__builtin_amdgcn_s_wait_asynccnt)
  __builtin_amdgcn_s_wait_asynccnt(0);
#else
  asm volatile("s_wait_asynccnt 0x0" ::: "memory");
#endif
}

__global__ __launch_bounds__(256) void vq_wmma_kernel(const float* __restrict__ x,
                                                      const float* __restrict__ cb,
                                                      float* __restrict__ out) {
  __shared__ float lds_cb[2][CHUNK * LDS_STRIDE];   // double-buffered codebook chunks
  __shared__ float lds_cn[CHUNK];
  __shared__ int   lds_idx[ROWS_PER_BLOCK];

  const int tid  = threadIdx.x;
  const int lane = tid & 31;
  const int wave = tid >> 5;
  const int col  = lane & 15;   // N-column selector (and M-row for A)
  const int half = lane >> 4;   // upper/lower 16-lane half

  const int r0    = blockIdx.x * ROWS_PER_BLOCK;  // flat row base, single batch per block
  const int bIdx  = r0 / SEQ;
  const int tBase = r0 - bIdx * SEQ;
  const float* xb = x + (size_t)bIdx * EMB_DIM * SEQ;

  // per-thread staging map: 8 x b128 per chunk, coalesced along channels
  const int stRow  = tid >> 5;          // base code row (advances by 8 per b128)
  const int stCoff = (tid & 31) << 2;   // channel offset (float index, 16B aligned)

  // ---- issue async stage of chunk 0 (overlaps with A preload below) ----
#pragma unroll
  for (int i = 0; i < 8; ++i) {
    const int row = stRow + i * 8;
    const uint64_t g = (uint64_t)(uintptr_t)(cb + (size_t)row * EMB_DIM + stCoff);
    const uint32_t l = (uint32_t)(uintptr_t)&lds_cb[0][row * LDS_STRIDE + stCoff];
    asm volatile("global_load_async_to_lds_b128 %0, %1, off" :: "v"(l), "v"(g) : "memory");
  }

  // ---- preload A (16 rows x 128 ch) into registers, WMMA 16x4 f32 A layout ----
  // lanes 0-15: K = 4s+0, 4s+1 ; lanes 16-31: K = 4s+2, 4s+3 ; M = lane%16
  const int tRow = tBase + wave * 16 + col;
  float a[64];
#pragma unroll
  for (int s = 0; s < 32; ++s) {
    const int c0 = 4 * s + 2 * half;
    a[2 * s + 0] = xb[(size_t)(c0 + 0) * SEQ + tRow];
    a[2 * s + 1] = xb[(size_t)(c0 + 1) * SEQ + tRow];
  }

  float bestD[8];
  int   bestI[8];
#pragma unroll
  for (int v = 0; v < 8; ++v) { bestD[v] = 3.402823466e38f; bestI[v] = 0; }

  for (int ch = 0; ch < NCHUNK; ++ch) {
    const int cur = ch & 1;
    // publish staged chunk `ch` to all waves; also guarantees every wave is done
    // computing on the other buffer before we overwrite it below
    wait_asynccnt0();
    __syncthreads();

    // issue async stage of next chunk into the other buffer (fire and forget)
    if (ch + 1 < NCHUNK) {
      const int nbase = (ch + 1) * CHUNK;
#pragma unroll
      for (int i = 0; i < 8; ++i) {
        const int row = stRow + i * 8;
        const uint64_t g =
            (uint64_t)(uintptr_t)(cb + (size_t)(nbase + row) * EMB_DIM + stCoff);
        const uint32_t l =
            (uint32_t)(uintptr_t)&lds_cb[cur ^ 1][row * LDS_STRIDE + stCoff];
        asm volatile("global_load_async_to_lds_b128 %0, %1, off" :: "v"(l), "v"(g) : "memory");
      }
    }

    // per-chunk half squared norms: 0.5*||c||^2
    if (tid < CHUNK) {
      float s = 0.f;
#pragma unroll 8
      for (int c = 0; c < EMB_DIM; ++c) {
        const float v = lds_cb[cur][tid * LDS_STRIDE + c];
        s += v * v;
      }
      lds_cn[tid] = 0.5f * s;
    }
    __syncthreads();

    const int cbase = ch * CHUNK;
    for (int jt = 0; jt < TILES_PER_CHUNK; ++jt) {
      const int   lrow    = jt * 16 + col;       // code row within chunk (B column N)
      const float hcn     = lds_cn[lrow];
      const int   codeIdx = cbase + lrow;
      const float* bp     = &lds_cb[cur][lrow * LDS_STRIDE + 2 * half];
      v8f acc = {0.f, 0.f, 0.f, 0.f, 0.f, 0.f, 0.f, 0.f};
#pragma unroll
      for (int s = 0; s < 32; ++s) {
        v2f av; av.x = a[2 * s]; av.y = a[2 * s + 1];
        const float2 bl = *(const float2*)(bp + 4 * s);   // ds_load_b64, conflict-free
        v2f bv; bv.x = bl.x; bv.y = bl.y;
        acc = __builtin_amdgcn_wmma_f32_16x16x4_f32(false, av, false, bv,
                                                    (short)0, acc, false, false);
      }
      // dist ~ 0.5||c||^2 - x.c   (per-row ||x||^2 dropped: constant under argmin)
#pragma unroll
      for (int v = 0; v < 8; ++v) {
        const float d = hcn - acc[v];
        if (d < bestD[v]) { bestD[v] = d; bestI[v] = codeIdx; }
      }
    }
  }

  // cross-lane argmin within each 16-lane half (rows M = v + 8*half)
#pragma unroll
  for (int v = 0; v < 8; ++v) {
#pragma unroll
    for (int m = 1; m <= 8; m <<= 1) {
      const float od = __shfl_xor(bestD[v], m, 32);
      const int   oi = __shfl_xor(bestI[v], m, 32);
      if (od < bestD[v] || (od == bestD[v] && oi < bestI[v])) {
        bestD[v] = od; bestI[v] = oi;
      }
    }
  }
  if (col == 0) {
#pragma unroll
    for (int v = 0; v < 8; ++v)
      lds_idx[wave * 16 + half * 8 + v] = bestI[v];
  }
  __syncthreads();

  // ---- output: z_q_x = x + (cb[idx]-x); z_q_x_bar = cb[idx]; [B,C,T] layout ----
  const size_t obase = (size_t)bIdx * EMB_DIM * SEQ;
#pragma unroll 4
  for (int it = 0; it < 64; ++it) {
    const int e    = it * 256 + tid;
    const int c    = e >> 7;       // channel
    const int rl   = e & 127;      // row within block (consecutive lanes -> consecutive t)
    const int code = lds_idx[rl];
    const size_t adr = obase + (size_t)c * SEQ + (size_t)(tBase + rl);
    const float xv = x[adr];
    const float cv = cb[(size_t)code * EMB_DIM + c];
    out[adr]       = xv + (cv - xv);   // straight-through value, same rounding as ref
    out[BCT + adr] = cv;
  }
}

extern "C" void kernel_launch(void* const* d_in, const int* in_sizes, int n_in,
                              void* d_out, int out_size, void* d_ws, size_t ws_size,
                              hipStream_t stream) {
  const float* z_e_x    = (const float*)d_in[0];
  const float* codebook = (const float*)d_in[1];
  float* out = (float*)d_out;
  vq_wmma_kernel<<<NROWS / ROWS_PER_BLOCK, 256, 0, stream>>>(z_e_x, codebook, out);
}